// FATAttention_82102594830398
// MI455X (gfx1250) — compile-verified
//
#include <hip/hip_runtime.h>
#include <math.h>

// Problem constants (from the reference)
constexpr int BB = 16;    // batch
constexpr int TT = 512;   // tokens
constexpr int DD = 256;   // model dim
constexpr int HH = 8;     // heads
constexpr int EE = 32;    // head dim
constexpr int HE = 256;   // H*E
constexpr int NB = 8;     // basis
constexpr int SINK_ITERS = 20;

typedef __attribute__((ext_vector_type(2))) float v2f;
typedef __attribute__((ext_vector_type(8))) float v8f;

__device__ __forceinline__ float wave_reduce_sum(float v) {
#pragma unroll
  for (int m = 16; m >= 1; m >>= 1) v += __shfl_xor(v, m, 32);
  return v;
}
__device__ __forceinline__ float wave_reduce_max(float v) {
#pragma unroll
  for (int m = 16; m >= 1; m >>= 1) v = fmaxf(v, __shfl_xor(v, m, 32));
  return v;
}

// -------- Kernel A: q_w[t,k] = <tok[t], Wq_meta[t,k]>, same for k_w --------
__global__ __launch_bounds__(256)
void meta_weights_kernel(const float* __restrict__ tok,
                         const float* __restrict__ Wq,
                         const float* __restrict__ Wk,
                         float* __restrict__ qw, float* __restrict__ kw) {
  const int t = blockIdx.x;
  const int tid = threadIdx.x;          // 256 threads == D
  const int lane = tid & 31, wave = tid >> 5;
  __shared__ float sq[NB][8];
  __shared__ float sk[NB][8];
  const float te = tok[t * DD + tid];
#pragma unroll
  for (int k2 = 0; k2 < NB; ++k2) {
    float vq = te * Wq[((size_t)t * NB + k2) * DD + tid];
    float vk = te * Wk[((size_t)t * NB + k2) * DD + tid];
    vq = wave_reduce_sum(vq);
    vk = wave_reduce_sum(vk);
    if (lane == 0) { sq[k2][wave] = vq; sk[k2][wave] = vk; }
  }
  __syncthreads();
  if (tid < NB) {
    float s = 0.f;
#pragma unroll
    for (int w = 0; w < 8; ++w) s += sq[tid][w];
    qw[t * NB + tid] = s;
  } else if (tid < 2 * NB) {
    const int k2 = tid - NB;
    float s = 0.f;
#pragma unroll
    for (int w = 0; w < 8; ++w) s += sk[k2][w];
    kw[t * NB + k2] = s;
  }
}

// -------- Kernel B: per-token projections via on-the-fly basis mixing --------
// querys[:,t,:] = X[:,t,:] @ (Σ_k q_w[t,k] Q_basis[k]);  same for keys; values use W_V[t]
__global__ __launch_bounds__(512)
void projections_kernel(const float* __restrict__ X,
                        const float* __restrict__ Qb,
                        const float* __restrict__ Kb,
                        const float* __restrict__ WV,
                        const float* __restrict__ qw,
                        const float* __restrict__ kw,
                        float* __restrict__ Qo, float* __restrict__ Ko,
                        float* __restrict__ Vo) {
  const int t = blockIdx.x;
  const int tid = threadIdx.x;                // 512 threads = 16 waves
  const int lane = tid & 31, wave = tid >> 5;
  __shared__ float Xs[BB][DD];                // 16 KB : A matrix (16 x 256)
  __shared__ float Bs[32][HE];                // 32 KB : mixed B slab (32 x 256)
  for (int i = tid; i < BB * DD; i += 512) {
    const int b = i / DD, d = i % DD;
    Xs[b][d] = X[((size_t)b * TT + t) * DD + d];
  }
  float mixq[NB], mixk[NB];
#pragma unroll
  for (int k2 = 0; k2 < NB; ++k2) {
    mixq[k2] = qw[t * NB + k2];
    mixk[k2] = kw[t * NB + k2];
  }
  const int koff = (lane >> 4) * 2;
  const int mrow = lane & 15;
  const int ncol = wave * 16 + (lane & 15);   // 16 waves cover 256 output cols
  for (int proj = 0; proj < 3; ++proj) {
    v8f acc = {0.f, 0.f, 0.f, 0.f, 0.f, 0.f, 0.f, 0.f};
    for (int slab = 0; slab < 8; ++slab) {
      const int d0 = slab * 32;
      __syncthreads();
      for (int i = tid; i < 32 * HE; i += 512) {
        const int r = i / HE, c = i % HE;
        const int d = d0 + r;
        float val;
        if (proj == 0) {
          val = 0.f;
#pragma unroll
          for (int k2 = 0; k2 < NB; ++k2)
            val += mixq[k2] * Qb[((size_t)k2 * DD + d) * HE + c];
        } else if (proj == 1) {
          val = 0.f;
#pragma unroll
          for (int k2 = 0; k2 < NB; ++k2)
            val += mixk[k2] * Kb[((size_t)k2 * DD + d) * HE + c];
        } else {
          val = WV[((size_t)t * DD + d) * HE + c];
        }
        Bs[r][c] = val;
      }
      __syncthreads();
#pragma unroll
      for (int ks = 0; ks < 8; ++ks) {
        v2f a, bf;
        a.x = Xs[mrow][d0 + ks * 4 + koff];
        a.y = Xs[mrow][d0 + ks * 4 + koff + 1];
        bf.x = Bs[ks * 4 + koff][ncol];
        bf.y = Bs[ks * 4 + koff + 1][ncol];
        acc = __builtin_amdgcn_wmma_f32_16x16x4_f32(false, a, false, bf,
                                                    (short)0, acc, false, false);
      }
    }
    float* outp = (proj == 0) ? Qo : (proj == 1) ? Ko : Vo;
#pragma unroll
    for (int i = 0; i < 8; ++i) {
      const int brow = i + 8 * (lane >> 4);   // C layout: M = vgpr + 8*(lane/16)
      outp[((size_t)brow * TT + t) * HE + ncol] = acc[i];
    }
  }
}

// -------- Kernel C: scores (q k^T / sqrt(E)) + row softmax -> P[hb,T,T] --------
__global__ __launch_bounds__(256)
void scores_softmax_kernel(const float* __restrict__ Qo,
                           const float* __restrict__ Ko,
                           float* __restrict__ P) {
  const int hb = blockIdx.y;
  const int h = hb / BB, b = hb % BB;
  const int row0 = blockIdx.x * 16;
  const int tid = threadIdx.x;                // 256 = 8 waves
  const int lane = tid & 31, wave = tid >> 5;
  __shared__ float Qs[16][EE];                // 2 KB
  __shared__ float Ss[16][TT];                // 32 KB
  for (int i = tid; i < 16 * EE; i += 256) {
    const int r = i / EE, e = i % EE;
    Qs[r][e] = Qo[((size_t)b * TT + row0 + r) * HE + h * EE + e];
  }
  __syncthreads();
  const int koff = (lane >> 4) * 2;
  const int mrow = lane & 15;
  const float scale = 0.17677669529663687f;   // 1/sqrt(32)
  for (int nt = wave; nt < TT / 16; nt += 8) {
    const int s0 = nt * 16;
    const int scol = s0 + (lane & 15);
    v8f acc = {0.f, 0.f, 0.f, 0.f, 0.f, 0.f, 0.f, 0.f};
#pragma unroll
    for (int ks = 0; ks < EE / 4; ++ks) {
      v2f a, bf;
      a.x = Qs[mrow][ks * 4 + koff];
      a.y = Qs[mrow][ks * 4 + koff + 1];
      // B[K=e][N=s] = keys[s][e]
      bf.x = Ko[((size_t)b * TT + scol) * HE + h * EE + ks * 4 + koff];
      bf.y = Ko[((size_t)b * TT + scol) * HE + h * EE + ks * 4 + koff + 1];
      acc = __builtin_amdgcn_wmma_f32_16x16x4_f32(false, a, false, bf,
                                                  (short)0, acc, false, false);
    }
#pragma unroll
    for (int i = 0; i < 8; ++i)
      Ss[i + 8 * (lane >> 4)][s0 + (lane & 15)] = acc[i] * scale;
  }
  __syncthreads();
  // softmax: each wave normalizes 2 rows of 512
  for (int rr = 0; rr < 2; ++rr) {
    const int r = wave * 2 + rr;
    float x[16];
    float m = -3.402823466e38f;
#pragma unroll
    for (int i = 0; i < 16; ++i) { x[i] = Ss[r][lane + 32 * i]; m = fmaxf(m, x[i]); }
    m = wave_reduce_max(m);
    float s = 0.f;
#pragma unroll
    for (int i = 0; i < 16; ++i) { x[i] = __expf(x[i] - m); s += x[i]; }
    s = wave_reduce_sum(s);
    const float inv = 1.0f / s;
#pragma unroll
    for (int i = 0; i < 16; ++i)
      P[((size_t)hb * TT + row0 + r) * TT + lane + 32 * i] = x[i] * inv;
  }
}

// -------- Kernel D: att = lam*A_sym + (1-lam)*att; att = 0.5*(att+att^T), in place --------
__global__ __launch_bounds__(256)
void mix_sym_kernel(const float* __restrict__ A,
                    const float* __restrict__ L,
                    float* __restrict__ P) {
  const int hb = blockIdx.y;
  const size_t base = (size_t)hb * TT * TT;
  for (int idx = blockIdx.x * blockDim.x + threadIdx.x; idx < TT * TT;
       idx += gridDim.x * blockDim.x) {
    const int t = idx / TT, s = idx % TT;
    if (t > s) continue;                      // each unordered pair handled once
    const float asym = 0.5f * (A[t * TT + s] + A[s * TT + t]);
    const float lts = 1.0f / (1.0f + __expf(-L[t * TT + s]));
    const float lst = 1.0f / (1.0f + __expf(-L[s * TT + t]));
    const float pts = P[base + (size_t)t * TT + s];
    const float pst = P[base + (size_t)s * TT + t];
    const float a1 = lts * asym + (1.f - lts) * pts;
    const float a2 = lst * asym + (1.f - lst) * pst;
    const float m = 0.5f * (a1 + a2);
    P[base + (size_t)t * TT + s] = m;
    P[base + (size_t)s * TT + t] = m;
  }
}

// -------- Kernel E: log-Sinkhorn via potentials u,v (LDS) + fused output GEMM --------
// u_t = LSE_s(M[t,s]-v_s); v_s = LSE_t(M[t,s]-u_t); out = exp(M-u-v) @ V
__global__ __launch_bounds__(1024)
void sinkhorn_out_kernel(const float* __restrict__ P,
                         const float* __restrict__ Vo,
                         float* __restrict__ out) {
  const int hb = blockIdx.x;
  const int h = hb / BB, b = hb % BB;
  const int tid = threadIdx.x;                // 1024 = 32 waves
  const int lane = tid & 31, wave = tid >> 5;
  __shared__ float u[TT];
  __shared__ float v[TT];
  const float* M = P + (size_t)hb * TT * TT;  // read-only from here on
  if (tid < TT) { u[tid] = 0.f; v[tid] = 0.f; }
  __syncthreads();
  for (int it = 0; it < SINK_ITERS; ++it) {
    // row update: 32 waves x 16 rows
    for (int r = wave * 16; r < wave * 16 + 16; ++r) {
      float x[16];
      float m = -3.402823466e38f;
#pragma unroll
      for (int i = 0; i < 16; ++i) {
        x[i] = M[(size_t)r * TT + lane + 32 * i] - v[lane + 32 * i];
        m = fmaxf(m, x[i]);
      }
      m = wave_reduce_max(m);
      float s = 0.f;
#pragma unroll
      for (int i = 0; i < 16; ++i) s += __expf(x[i] - m);
      s = wave_reduce_sum(s);
      if (lane == 0) u[r] = m + __logf(s);
    }
    __syncthreads();
    // col update: waves 0..15 own 32 columns each, online LSE down t
    if (wave < 16) {
      const int sc = wave * 32 + lane;
      float m = -3.402823466e38f, s = 0.f;
      for (int tt2 = 0; tt2 < TT; ++tt2) {
        const float x = M[(size_t)tt2 * TT + sc] - u[tt2];
        const float mn = fmaxf(m, x);
        s = s * __expf(m - mn) + __expf(x - mn);
        m = mn;
      }
      v[sc] = m + __logf(s);
    }
    __syncthreads();
  }
  // output GEMM: out[t, h*E+e] = sum_s exp(M[t,s]-u_t-v_s) * values[b,s,h*E+e]
  const int m0 = wave * 16;                   // 32 waves cover T=512 rows
  const int koff = (lane >> 4) * 2;
  const int mrow = m0 + (lane & 15);
  const int n = lane & 15;
  v8f acc0 = {0.f, 0.f, 0.f, 0.f, 0.f, 0.f, 0.f, 0.f};
  v8f acc1 = {0.f, 0.f, 0.f, 0.f, 0.f, 0.f, 0.f, 0.f};
  for (int kt = 0; kt < TT / 4; ++kt) {
    const int k0 = kt * 4 + koff;
    v2f a, b0, b1;
    a.x = __expf(M[(size_t)mrow * TT + k0] - u[mrow] - v[k0]);
    a.y = __expf(M[(size_t)mrow * TT + k0 + 1] - u[mrow] - v[k0 + 1]);
    const float* vp0 = Vo + ((size_t)b * TT + k0) * HE + h * EE;
    const float* vp1 = Vo + ((size_t)b * TT + k0 + 1) * HE + h * EE;
    b0.x = vp0[n];      b0.y = vp1[n];
    b1.x = vp0[n + 16]; b1.y = vp1[n + 16];
    acc0 = __builtin_amdgcn_wmma_f32_16x16x4_f32(false, a, false, b0,
                                                 (short)0, acc0, false, false);
    acc1 = __builtin_amdgcn_wmma_f32_16x16x4_f32(false, a, false, b1,
                                                 (short)0, acc1, false, false);
  }
#pragma unroll
  for (int i = 0; i < 8; ++i) {
    const int trow = m0 + i + 8 * (lane >> 4);
    out[((size_t)b * TT + trow) * DD + h * EE + n] = acc0[i];
    out[((size_t)b * TT + trow) * DD + h * EE + n + 16] = acc1[i];
  }
}

extern "C" void kernel_launch(void* const* d_in, const int* in_sizes, int n_in,
                              void* d_out, int out_size, void* d_ws, size_t ws_size,
                              hipStream_t stream) {
  (void)in_sizes; (void)n_in; (void)out_size; (void)ws_size;
  const float* X   = (const float*)d_in[0];
  const float* tok = (const float*)d_in[1];
  const float* Wq  = (const float*)d_in[2];
  const float* Wk  = (const float*)d_in[3];
  const float* Qb  = (const float*)d_in[4];
  const float* Kb  = (const float*)d_in[5];
  const float* WV  = (const float*)d_in[6];
  const float* A   = (const float*)d_in[7];
  const float* L   = (const float*)d_in[8];
  float* out = (float*)d_out;

  float* ws = (float*)d_ws;
  float* qw = ws;                       ws += TT * NB;
  float* kw = ws;                       ws += TT * NB;
  float* Qo = ws;                       ws += (size_t)BB * TT * HE;
  float* Ko = ws;                       ws += (size_t)BB * TT * HE;
  float* Vo = ws;                       ws += (size_t)BB * TT * HE;
  float* P  = ws;                       // H*B*T*T floats (134 MB)

  meta_weights_kernel<<<TT, 256, 0, stream>>>(tok, Wq, Wk, qw, kw);
  projections_kernel<<<TT, 512, 0, stream>>>(X, Qb, Kb, WV, qw, kw, Qo, Ko, Vo);
  scores_softmax_kernel<<<dim3(TT / 16, HH * BB), 256, 0, stream>>>(Qo, Ko, P);
  mix_sym_kernel<<<dim3(256, HH * BB), 256, 0, stream>>>(A, L, P);
  sinkhorn_out_kernel<<<HH * BB, 1024, 0, stream>>>(P, Vo, out);
}